// HeteroSAGEConvLayer_1099511628137
// MI455X (gfx1250) — compile-verified
//
#include <hip/hip_runtime.h>
#include <hip/hip_bf16.h>

typedef float v2f __attribute__((ext_vector_type(2)));
typedef float v8f __attribute__((ext_vector_type(8)));

#define HID 128

// ---------------------------------------------------------------------------
// Y[M,HID] = X[M,HID] @ W^T + bias        (W: [HID,HID] row-major, torch Linear)
// One 256-thread block = 8 waves; block covers rows [16*blockIdx.x, +16),
// wave w covers output cols [16*w, +16). K=128 -> 32 x V_WMMA_F32_16X16X4_F32.
// ---------------------------------------------------------------------------
__global__ void __launch_bounds__(256) gemm_node(const float* __restrict__ X,
                                                 const float* __restrict__ W,
                                                 const float* __restrict__ bias,
                                                 float* __restrict__ Y) {
  const int lane  = threadIdx.x & 31;
  const int wave  = threadIdx.x >> 5;          // 0..7 -> N tile
  const int m0    = blockIdx.x << 4;
  const int n0    = wave << 4;
  const int l15   = lane & 15;
  const int hi    = lane >> 4;                 // 0 or 1
  const int khalf = hi << 1;                   // 0 or 2

  const float* xp = X + (size_t)(m0 + l15) * HID + khalf;   // A frag: X[row][4k+khalf..+1]
  const float* wp = W + (size_t)(n0 + l15) * HID + khalf;   // B frag: W[col][4k+khalf..+1]

  v8f acc = {};
#pragma unroll 4
  for (int kk = 0; kk < 32; ++kk) {
    v2f a = *(const v2f*)(xp + (kk << 2));
    v2f b = *(const v2f*)(wp + (kk << 2));
    acc = __builtin_amdgcn_wmma_f32_16x16x4_f32(false, a, false, b, (short)0, acc,
                                                false, false);
  }

  const int   col = n0 + l15;
  const float bb  = bias[col];
  float* yp = Y + (size_t)(m0 + (hi << 3)) * HID + col;     // row = m0 + v + 8*hi
#pragma unroll
  for (int v = 0; v < 8; ++v)
    yp[(size_t)v * HID] = acc[v] + bb;
}

// ---------------------------------------------------------------------------
// Y[M,HID] = relu( X1 @ W[:, :128]^T + X2 @ W[:, 128:]^T + bias )
// W: [HID, 2*HID] row-major (torch Linear on concat([X1,X2])).
// ---------------------------------------------------------------------------
__global__ void __launch_bounds__(256) gemm_concat_relu(const float* __restrict__ X1,
                                                        const float* __restrict__ X2,
                                                        const float* __restrict__ W,
                                                        const float* __restrict__ bias,
                                                        float* __restrict__ Y) {
  const int lane  = threadIdx.x & 31;
  const int wave  = threadIdx.x >> 5;
  const int m0    = blockIdx.x << 4;
  const int n0    = wave << 4;
  const int l15   = lane & 15;
  const int hi    = lane >> 4;
  const int khalf = hi << 1;

  const float* x1 = X1 + (size_t)(m0 + l15) * HID + khalf;
  const float* x2 = X2 + (size_t)(m0 + l15) * HID + khalf;
  const float* w1 = W + (size_t)(n0 + l15) * (2 * HID) + khalf;
  const float* w2 = w1 + HID;

  v8f acc = {};
#pragma unroll 4
  for (int kk = 0; kk < 32; ++kk) {
    v2f a = *(const v2f*)(x1 + (kk << 2));
    v2f b = *(const v2f*)(w1 + (kk << 2));
    acc = __builtin_amdgcn_wmma_f32_16x16x4_f32(false, a, false, b, (short)0, acc,
                                                false, false);
  }
#pragma unroll 4
  for (int kk = 0; kk < 32; ++kk) {
    v2f a = *(const v2f*)(x2 + (kk << 2));
    v2f b = *(const v2f*)(w2 + (kk << 2));
    acc = __builtin_amdgcn_wmma_f32_16x16x4_f32(false, a, false, b, (short)0, acc,
                                                false, false);
  }

  const int   col = n0 + l15;
  const float bb  = bias[col];
  float* yp = Y + (size_t)(m0 + (hi << 3)) * HID + col;
#pragma unroll
  for (int v = 0; v < 8; ++v)
    yp[(size_t)v * HID] = fmaxf(acc[v] + bb, 0.0f);
}

// ---------------------------------------------------------------------------
// Per edge: sum[dst] += h[src] (128 f32 atomics per edge, 4 per lane),
// cnt[dst] += 1. One wave per edge; float4 gather per lane.
// ei layout: ei[0..nE) = src, ei[nE..2nE) = dst.
// ---------------------------------------------------------------------------
__global__ void __launch_bounds__(256) scatter_edges(const float* __restrict__ h,
                                                     const int* __restrict__ ei,
                                                     float* __restrict__ sum,
                                                     float* __restrict__ cnt,
                                                     int nE) {
  const int e = (blockIdx.x << 3) + (threadIdx.x >> 5);
  if (e >= nE) return;
  const int lane = threadIdx.x & 31;
  const int s = ei[e];
  const int d = ei[nE + e];
  const float4 v = *(const float4*)(h + (size_t)s * HID + (lane << 2));
  float* p = sum + (size_t)d * HID + (lane << 2);
  atomicAdd(p + 0, v.x);
  atomicAdd(p + 1, v.y);
  atomicAdd(p + 2, v.z);
  atomicAdd(p + 3, v.w);
  if (lane == 0) atomicAdd(cnt + d, 1.0f);
}

// neigh_p (in-place into sum_c): 0.5*(sum_c/max(cnt_c,1) + sum_w/max(cnt_w,1))
__global__ void __launch_bounds__(256) finalize_paper(float* __restrict__ sum_c,
                                                      const float* __restrict__ sum_w,
                                                      const float* __restrict__ cnt_c,
                                                      const float* __restrict__ cnt_w,
                                                      int total) {
  int i = blockIdx.x * 256 + threadIdx.x;
  if (i >= total) return;
  int node = i >> 7;
  float ic = 1.0f / fmaxf(cnt_c[node], 1.0f);
  float iw = 1.0f / fmaxf(cnt_w[node], 1.0f);
  sum_c[i] = 0.5f * (sum_c[i] * ic + sum_w[i] * iw);
}

// in-place: sum /= max(cnt,1)
__global__ void __launch_bounds__(256) finalize_div(float* __restrict__ sum,
                                                    const float* __restrict__ cnt,
                                                    int total) {
  int i = blockIdx.x * 256 + threadIdx.x;
  if (i >= total) return;
  int node = i >> 7;
  sum[i] = sum[i] / fmaxf(cnt[node], 1.0f);
}

extern "C" void kernel_launch(void* const* d_in, const int* in_sizes, int n_in,
                              void* d_out, int out_size, void* d_ws, size_t ws_size,
                              hipStream_t stream) {
  // inputs, setup_inputs() order:
  const float* x_paper   = (const float*)d_in[0];
  const float* x_author  = (const float*)d_in[1];
  const int*   ei_cites  = (const int*)d_in[2];
  const int*   ei_writes = (const int*)d_in[3];
  const int*   ei_written= (const int*)d_in[4];
  const float* W_cites   = (const float*)d_in[5];
  const float* b_cites   = (const float*)d_in[6];
  const float* W_writes  = (const float*)d_in[7];
  const float* b_writes  = (const float*)d_in[8];
  const float* W_written = (const float*)d_in[9];
  const float* b_written = (const float*)d_in[10];
  const float* Ws_paper  = (const float*)d_in[11];
  const float* bs_paper  = (const float*)d_in[12];
  const float* Ws_author = (const float*)d_in[13];
  const float* bs_author = (const float*)d_in[14];
  const float* Wc_paper  = (const float*)d_in[15];
  const float* bc_paper  = (const float*)d_in[16];
  const float* Wc_author = (const float*)d_in[17];
  const float* bc_author = (const float*)d_in[18];

  const int NP = in_sizes[0] / HID;   // 100000
  const int NA = in_sizes[1] / HID;   // 50000
  const int NE = in_sizes[2] / 2;     // 500000
  const size_t NPH = (size_t)NP * HID;
  const size_t NAH = (size_t)NA * HID;

  // workspace carve-up (f32). sums+counts are contiguous so one memset covers them.
  float* ws      = (float*)d_ws;
  float* h_c     = ws;                 // NPH  : x_paper  @ W_cites^T  + b
  float* h_w     = h_c  + NPH;         // NAH  : x_author @ W_writes^T + b
  float* h_wr    = h_w  + NAH;         // NPH  : x_paper  @ W_written^T+ b
  float* sum_c   = h_wr + NPH;         // NPH  ---+
  float* sum_w   = sum_c + NPH;        // NPH     | zeroed each call
  float* sum_wr  = sum_w + NPH;        // NAH     |
  float* cnt_c   = sum_wr + NAH;       // NP      |
  float* cnt_w   = cnt_c + NP;         // NP      |
  float* cnt_wr  = cnt_w + NP;         // NA   ---+
  float* self_p  = cnt_wr + NA;        // NPH
  float* self_a  = self_p + NPH;       // NAH

  const size_t zero_bytes = (2 * NPH + NAH + 2 * (size_t)NP + (size_t)NA) * sizeof(float);
  hipMemsetAsync(sum_c, 0, zero_bytes, stream);

  float* out_p = (float*)d_out;        // [NP, HID]
  float* out_a = out_p + NPH;          // [NA, HID]

  const int gP = NP / 16;              // 6250 (exact)
  const int gA = NA / 16;              // 3125 (exact)
  const int gE = (NE + 7) / 8;

  // node transforms (Linear commutes with the gather)
  gemm_node<<<gP, 256, 0, stream>>>(x_paper,  W_cites,   b_cites,   h_c);
  gemm_node<<<gA, 256, 0, stream>>>(x_author, W_writes,  b_writes,  h_w);
  gemm_node<<<gP, 256, 0, stream>>>(x_paper,  W_written, b_written, h_wr);
  gemm_node<<<gP, 256, 0, stream>>>(x_paper,  Ws_paper,  bs_paper,  self_p);
  gemm_node<<<gA, 256, 0, stream>>>(x_author, Ws_author, bs_author, self_a);

  // scatter-sum per edge type
  scatter_edges<<<gE, 256, 0, stream>>>(h_c,  ei_cites,   sum_c,  cnt_c,  NE);
  scatter_edges<<<gE, 256, 0, stream>>>(h_w,  ei_writes,  sum_w,  cnt_w,  NE);
  scatter_edges<<<gE, 256, 0, stream>>>(h_wr, ei_written, sum_wr, cnt_wr, NE);

  // means: neigh_p = 0.5*(mean_c + mean_w) (into sum_c); agg_written = mean (in place)
  finalize_paper<<<(int)((NPH + 255) / 256), 256, 0, stream>>>(sum_c, sum_w, cnt_c, cnt_w,
                                                               (int)NPH);
  finalize_div<<<(int)((NAH + 255) / 256), 256, 0, stream>>>(sum_wr, cnt_wr, (int)NAH);

  // fused concat + Linear + ReLU straight into d_out
  gemm_concat_relu<<<gP, 256, 0, stream>>>(self_p, sum_c,  Wc_paper,  bc_paper,  out_p);
  gemm_concat_relu<<<gA, 256, 0, stream>>>(self_a, sum_wr, Wc_author, bc_author, out_a);
}